// StateOuterProductCumsumReadout_83614423318659
// MI455X (gfx1250) — compile-verified
//
#include <hip/hip_runtime.h>

// CDNA5 / gfx1250, wave32. All matrix math uses V_WMMA_F32_16X16X4_F32 to keep
// exact f32 precision (problem is bandwidth-bound at this size, bf16 buys nothing).

typedef float v2f __attribute__((ext_vector_type(2)));
typedef float v8f __attribute__((ext_vector_type(8)));

#define T_TOT  2048
#define B_SZ   4
#define D_SZ   256
#define N_SZ   256
#define C_SZ   128
#define NCHUNK 16
#define EPS_F  1e-8f

__device__ __forceinline__ v8f wmma_f32_k4(v2f a, v2f b, v8f c) {
  // D = A(16x4) * B(4x16) + C, f32 throughout.
  return __builtin_amdgcn_wmma_f32_16x16x4_f32(false, a, false, b, (short)0, c, false, false);
}

// ---------------------------------------------------------------------------
// Kernel 1: per (chunk,batch) compute kd[t,n] = k * cp[t]/(cp[last]+eps) and
// prefix[n] = clip(alpha[0]).  One thread per n column, 256 threads/block.
// ---------------------------------------------------------------------------
__global__ void k_decay(const float* __restrict__ kin, const float* __restrict__ alpha,
                        float* __restrict__ kd, float* __restrict__ prefix) {
  const int c = blockIdx.x / B_SZ;
  const int b = blockIdx.x % B_SZ;
  const int n = threadIdx.x;                       // 0..255
  const long stride = (long)B_SZ * N_SZ;           // between consecutive t
  const long base = ((long)c * C_SZ * B_SZ + b) * N_SZ + n;

  float P = 1.0f;
  #pragma unroll 4
  for (int t = 0; t < C_SZ; ++t) {
    float a = alpha[base + (long)t * stride];
    a = (a < EPS_F) ? EPS_F : a;
    P *= a;
  }
  const float inv = 1.0f / (P + EPS_F);
  float r = 1.0f;
  #pragma unroll 4
  for (int t = 0; t < C_SZ; ++t) {
    const long idx = base + (long)t * stride;
    float a = alpha[idx];
    a = (a < EPS_F) ? EPS_F : a;
    r *= a;
    kd[idx] = kin[idx] * (r * inv);
  }
  float a0 = alpha[base];
  a0 = (a0 < EPS_F) ? EPS_F : a0;
  prefix[(c * B_SZ + b) * N_SZ + n] = a0;
}

// ---------------------------------------------------------------------------
// Kernel 2: U[c,b,d,n] = sum_t v[t,d] * kd[t,n].   GEMM M=256 N=256 K=128.
// Grid (B*NC, D/64, N/64), 128 threads = 4 waves, 64x64 tile per block.
// ---------------------------------------------------------------------------
__global__ void k_gemmU(const float* __restrict__ v, const float* __restrict__ kd,
                        float* __restrict__ U) {
  __shared__ float Av[C_SZ][65];   // Av[t][dLocal]  (A = v^T, read as Av[k][m])
  __shared__ float Bk[C_SZ][65];   // Bk[t][nLocal]
  const int c = blockIdx.x / B_SZ, b = blockIdx.x % B_SZ;
  const int dBase = blockIdx.y * 64, nBase = blockIdx.z * 64;
  const int tid = threadIdx.x, lane = tid & 31, w = tid >> 5;

  for (int idx = tid; idx < C_SZ * 64; idx += 128) {
    const int t = idx >> 6, i = idx & 63;
    const long row = (long)(c * C_SZ + t) * B_SZ + b;
    Av[t][i] = v[row * D_SZ + dBase + i];
    Bk[t][i] = kd[row * N_SZ + nBase + i];
  }
  __syncthreads();

  const int half = lane >> 4, lan = lane & 15;
  const int mLoc = w * 16 + lan;                   // wave owns 16 d-rows
  v8f acc[4] = {};
  #pragma unroll
  for (int kk = 0; kk < C_SZ; kk += 4) {
    const int ka = kk + half * 2;
    v2f a; a.x = Av[ka][mLoc]; a.y = Av[ka + 1][mLoc];
    #pragma unroll
    for (int j = 0; j < 4; ++j) {
      v2f bf; bf.x = Bk[ka][j * 16 + lan]; bf.y = Bk[ka + 1][j * 16 + lan];
      acc[j] = wmma_f32_k4(a, bf, acc[j]);
    }
  }
  const long ub = (long)(c * B_SZ + b) * D_SZ;
  #pragma unroll
  for (int j = 0; j < 4; ++j)
    #pragma unroll
    for (int i = 0; i < 8; ++i) {
      const int row = dBase + w * 16 + half * 8 + i;
      const int col = nBase + j * 16 + lan;
      U[(ub + row) * N_SZ + col] = acc[j][i];
    }
}

// ---------------------------------------------------------------------------
// Kernel 3: per (chunk,batch): A = q * kd^T (128x128, K=256), causal mask
// (s <= t), then y2 = A * v (128x256, K=128) written to out.
// 256 threads = 8 waves; wave w owns t-rows [16w,16w+16).
// ---------------------------------------------------------------------------
__global__ void k_intra(const float* __restrict__ q, const float* __restrict__ kd,
                        const float* __restrict__ v, float* __restrict__ out) {
  __shared__ float As[C_SZ][C_SZ + 4];   // masked attention matrix
  __shared__ float Pa[C_SZ][65];         // q panel, reused for v panel
  __shared__ float Pb[C_SZ][65];         // kd panel
  const int c = blockIdx.x / B_SZ, b = blockIdx.x % B_SZ;
  const int tid = threadIdx.x, lane = tid & 31, w = tid >> 5;
  const int half = lane >> 4, lan = lane & 15;
  const int mRow = w * 16 + lan;

  // ---- phase 1: A[t,s] = sum_n q[t,n] kd[s,n] ----
  v8f acc[8] = {};
  for (int kn = 0; kn < N_SZ; kn += 64) {
    for (int idx = tid; idx < C_SZ * 64; idx += 256) {
      const int t = idx >> 6, j = idx & 63;
      const long row = ((long)(c * C_SZ + t) * B_SZ + b) * N_SZ + kn + j;
      Pa[t][j] = q[row];
      Pb[t][j] = kd[row];
    }
    __syncthreads();
    #pragma unroll
    for (int kk = 0; kk < 64; kk += 4) {
      const int ka = kk + half * 2;
      v2f a; a.x = Pa[mRow][ka]; a.y = Pa[mRow][ka + 1];
      #pragma unroll
      for (int j = 0; j < 8; ++j) {
        // B(k, s) = kd[s][k]  -> transposed read from padded LDS tile
        v2f bf; bf.x = Pb[j * 16 + lan][ka]; bf.y = Pb[j * 16 + lan][ka + 1];
        acc[j] = wmma_f32_k4(a, bf, acc[j]);
      }
    }
    __syncthreads();
  }
  // causal mask (inclusive) and stash into LDS
  #pragma unroll
  for (int j = 0; j < 8; ++j)
    #pragma unroll
    for (int i = 0; i < 8; ++i) {
      const int row = w * 16 + half * 8 + i;   // t
      const int col = j * 16 + lan;            // s
      As[row][col] = (col <= row) ? acc[j][i] : 0.0f;
    }
  __syncthreads();

  // ---- phase 2: y2 = As * v ----
  for (int dp = 0; dp < D_SZ; dp += 64) {
    for (int idx = tid; idx < C_SZ * 64; idx += 256) {
      const int s = idx >> 6, j = idx & 63;
      Pa[s][j] = v[((long)(c * C_SZ + s) * B_SZ + b) * D_SZ + dp + j];
    }
    __syncthreads();
    v8f acc2[4] = {};
    #pragma unroll
    for (int kk = 0; kk < C_SZ; kk += 4) {
      const int ka = kk + half * 2;
      v2f a; a.x = As[mRow][ka]; a.y = As[mRow][ka + 1];
      #pragma unroll
      for (int j = 0; j < 4; ++j) {
        v2f bf; bf.x = Pa[ka][j * 16 + lan]; bf.y = Pa[ka + 1][j * 16 + lan];
        acc2[j] = wmma_f32_k4(a, bf, acc2[j]);
      }
    }
    #pragma unroll
    for (int j = 0; j < 4; ++j)
      #pragma unroll
      for (int i = 0; i < 8; ++i) {
        const int row = w * 16 + half * 8 + i;
        const int col = dp + j * 16 + lan;
        out[((long)(c * C_SZ + row) * B_SZ + b) * D_SZ + col] = acc2[j][i];
      }
    __syncthreads();
  }
}

// ---------------------------------------------------------------------------
// Kernel 4: elementwise chunk scan.  Sst[c] = state BEFORE chunk c.
// s_{c} = s_{c-1} * prefix_{c-1} + U_{c-1},  s_0 = 0.  One thread per (b,d,n).
// ---------------------------------------------------------------------------
__global__ void k_scan(const float* __restrict__ U, const float* __restrict__ prefix,
                       float* __restrict__ Sst) {
  const long g = (long)blockIdx.x * blockDim.x + threadIdx.x;  // over B*D*N
  const int n = (int)(g % N_SZ);
  const int b = (int)(g / ((long)D_SZ * N_SZ));
  const long cs = (long)B_SZ * D_SZ * N_SZ;
  float s = 0.0f;
  for (int c = 0; c < NCHUNK; ++c) {
    Sst[c * cs + g] = s;
    s = s * prefix[(c * B_SZ + b) * N_SZ + n] + U[c * cs + g];
  }
}

// ---------------------------------------------------------------------------
// Kernel 5: y1[t,d] = sum_n (q[t,n]*prefix[n]) * S_prev[d,n], accumulated into
// out.  Grid (B*NC, D/64), 256 threads = 8 waves.
// ---------------------------------------------------------------------------
__global__ void k_inter(const float* __restrict__ q, const float* __restrict__ prefix,
                        const float* __restrict__ Sst, float* __restrict__ out) {
  __shared__ float Qp[C_SZ][65];
  __shared__ float Sp[64][65];
  const int c = blockIdx.x / B_SZ, b = blockIdx.x % B_SZ;
  const int dBase = blockIdx.y * 64;
  const int tid = threadIdx.x, lane = tid & 31, w = tid >> 5;
  const int half = lane >> 4, lan = lane & 15;
  const int mRow = w * 16 + lan;
  const long cs = (long)B_SZ * D_SZ * N_SZ;

  v8f acc[4] = {};
  for (int np = 0; np < N_SZ; np += 64) {
    for (int idx = tid; idx < C_SZ * 64; idx += 256) {
      const int t = idx >> 6, j = idx & 63;
      const float pf = prefix[(c * B_SZ + b) * N_SZ + np + j];
      Qp[t][j] = q[((long)(c * C_SZ + t) * B_SZ + b) * N_SZ + np + j] * pf;
    }
    for (int idx = tid; idx < 64 * 64; idx += 256) {
      const int i = idx >> 6, j = idx & 63;
      Sp[i][j] = Sst[(long)c * cs + ((long)b * D_SZ + dBase + i) * N_SZ + np + j];
    }
    __syncthreads();
    #pragma unroll
    for (int kk = 0; kk < 64; kk += 4) {
      const int ka = kk + half * 2;
      v2f a; a.x = Qp[mRow][ka]; a.y = Qp[mRow][ka + 1];
      #pragma unroll
      for (int j = 0; j < 4; ++j) {
        // B(k=n, col=d) = S[d][n] -> transposed read from padded LDS tile
        v2f bf; bf.x = Sp[j * 16 + lan][ka]; bf.y = Sp[j * 16 + lan][ka + 1];
        acc[j] = wmma_f32_k4(a, bf, acc[j]);
      }
    }
    __syncthreads();
  }
  #pragma unroll
  for (int j = 0; j < 4; ++j)
    #pragma unroll
    for (int i = 0; i < 8; ++i) {
      const int row = w * 16 + half * 8 + i;
      const int col = dBase + j * 16 + lan;
      out[((long)(c * C_SZ + row) * B_SZ + b) * D_SZ + col] += acc[j][i];
    }
}

// ---------------------------------------------------------------------------
extern "C" void kernel_launch(void* const* d_in, const int* in_sizes, int n_in,
                              void* d_out, int out_size, void* d_ws, size_t ws_size,
                              hipStream_t stream) {
  const float* v     = (const float*)d_in[0];
  const float* k     = (const float*)d_in[1];
  const float* alpha = (const float*)d_in[2];
  const float* q     = (const float*)d_in[3];
  float* out = (float*)d_out;
  float* ws  = (float*)d_ws;

  const long KD_N = (long)T_TOT * B_SZ * N_SZ;          // 2,097,152 floats
  const long PF_N = (long)NCHUNK * B_SZ * N_SZ;         // 16,384
  const long U_N  = (long)NCHUNK * B_SZ * D_SZ * N_SZ;  // 4,194,304
  float* kd     = ws;
  float* prefix = kd + KD_N;
  float* U      = prefix + PF_N;
  float* Sst    = U + U_N;                              // total ~42 MB

  k_decay<<<dim3(B_SZ * NCHUNK), 256, 0, stream>>>(k, alpha, kd, prefix);
  k_gemmU<<<dim3(B_SZ * NCHUNK, D_SZ / 64, N_SZ / 64), 128, 0, stream>>>(v, kd, U);
  k_intra<<<dim3(B_SZ * NCHUNK), 256, 0, stream>>>(q, kd, v, out);
  k_scan<<<dim3((B_SZ * D_SZ * N_SZ) / 256), 256, 0, stream>>>(U, prefix, Sst);
  k_inter<<<dim3(B_SZ * NCHUNK, D_SZ / 64), 256, 0, stream>>>(q, prefix, Sst, out);
}